// MoEBase_88811333746698
// MI455X (gfx1250) — compile-verified
//
#include <hip/hip_runtime.h>
#include <hip/hip_bf16.h>

// ---------------- problem constants ----------------
#define kD 1024     // d_model
#define kH 1024     // hidden
#define kE 8        // experts
#define kK 2        // top-k
#define kN 4096     // tokens (B*S)
#define kSLOTS 3    // K routed + 1 shared slot per token

typedef __attribute__((ext_vector_type(2))) float v2f;
typedef __attribute__((ext_vector_type(4))) float v4f;
typedef __attribute__((ext_vector_type(8))) float v8f;

// ---------------- workspace layout (bytes) ----------------
#define OFF_COEF   ((size_t)0)                       // float[kN*3]            49152
#define OFF_LIST   ((size_t)49152)                   // int[(kE+1)*kN]        147456
#define OFF_CNT    ((size_t)196608)                  // int[16]
#define OFF_PROBS  ((size_t)196864)                  // float[kN*kE]          131072
#define OFF_LSE2   ((size_t)327936)                  // float[kN]              16384
#define OFF_HBUF   ((size_t)344320)                  // float[kN*3*kH]      50331648
#define OFF_OBUF   ((size_t)50675968)                // float[kN*3*kD]      50331648

__device__ __forceinline__ v8f wmma4(v2f a, v2f b, v8f c) {
  // V_WMMA_F32_16X16X4_F32 : D = A(16x4,f32) * B(4x16,f32) + C(16x16,f32)
  return __builtin_amdgcn_wmma_f32_16x16x4_f32(
      false, a, false, b, (short)0, c, false, false);
}

// Async DMA: global memory -> LDS (16B per lane), tracked by ASYNCcnt.
__device__ __forceinline__ void async_copy_b128(unsigned int lds, const float* g) {
  asm volatile("global_load_async_to_lds_b128 %0, %1, off"
               :: "v"(lds), "v"(g) : "memory");
}
__device__ __forceinline__ void wait_async_le(void) {}
#define WAIT_ASYNC(n) asm volatile("s_wait_asynccnt " #n ::: "memory")

__device__ __forceinline__ unsigned int lds_addr(const void* p) {
  return (unsigned int)(unsigned long long)p;   // low 32 bits = LDS offset
}

// ---------------- 0: init expert counters ----------------
__global__ void moe_init_counts(int* __restrict__ cnt) {
  int i = threadIdx.x;
  if (i < kE) cnt[i] = 0;
  else if (i == kE) cnt[kE] = kN;   // shared expert serves all tokens
}

// ---------------- 1: router (1 wave / token) ----------------
__global__ __launch_bounds__(256) void moe_router(
    const float* __restrict__ x, const float* __restrict__ wr,
    int* __restrict__ list, int* __restrict__ cnt,
    float* __restrict__ coef, float* __restrict__ probsBuf,
    float* __restrict__ lse2Buf) {
  int wave = threadIdx.x >> 5;
  int lane = threadIdx.x & 31;
  int t = blockIdx.x * 8 + wave;

  float lg[kE];
#pragma unroll
  for (int e = 0; e < kE; ++e) lg[e] = 0.0f;

  for (int d = lane; d < kD; d += 32) {
    float xv = x[(size_t)t * kD + d];
    const v4f* wrow = (const v4f*)(wr + (size_t)d * kE);
    v4f w0 = wrow[0], w1 = wrow[1];
    lg[0] += xv * w0.x; lg[1] += xv * w0.y; lg[2] += xv * w0.z; lg[3] += xv * w0.w;
    lg[4] += xv * w1.x; lg[5] += xv * w1.y; lg[6] += xv * w1.z; lg[7] += xv * w1.w;
  }
#pragma unroll
  for (int off = 16; off > 0; off >>= 1)
#pragma unroll
    for (int e = 0; e < kE; ++e) lg[e] += __shfl_xor(lg[e], off, 32);

  float mx = lg[0];
#pragma unroll
  for (int e = 1; e < kE; ++e) mx = fmaxf(mx, lg[e]);
  float pr[kE]; float se = 0.0f;
#pragma unroll
  for (int e = 0; e < kE; ++e) { pr[e] = __expf(lg[e] - mx); se += pr[e]; }
  float inv = 1.0f / se;
#pragma unroll
  for (int e = 0; e < kE; ++e) pr[e] *= inv;
  float lse = mx + __logf(se);

  int i0 = 0;
#pragma unroll
  for (int e = 1; e < kE; ++e) if (pr[e] > pr[i0]) i0 = e;
  int i1 = (i0 == 0) ? 1 : 0;
#pragma unroll
  for (int e = 0; e < kE; ++e) if (e != i0 && pr[e] > pr[i1]) i1 = e;

  if (lane == 0) {
#pragma unroll
    for (int e = 0; e < kE; ++e) probsBuf[(size_t)t * kE + e] = pr[e];
    lse2Buf[t] = lse * lse;
    const float rscale = (float)kK / (float)(kK + 1);   // 2/3
    int p0 = atomicAdd(&cnt[i0], 1);
    list[i0 * kN + p0] = t * kSLOTS + 0;
    coef[t * kSLOTS + 0] = rscale * pr[i0];
    int p1 = atomicAdd(&cnt[i1], 1);
    list[i1 * kN + p1] = t * kSLOTS + 1;
    coef[t * kSLOTS + 1] = rscale * pr[i1];
    list[kE * kN + t] = t * kSLOTS + 2;                 // shared slot
    coef[t * kSLOTS + 2] = 1.0f / (float)(kK + 1);      // 1/3
  }
}

// ---------------- 2: gemm1  h = silu(Xe@W1) * (Xe@W3) ----------------
#define BM 64
#define BN 64
#define BK 32
#define XPAD 36
#define NT (kD / BK)   // 32 k-chunks

__global__ __launch_bounds__(256) void moe_gemm1(
    const float* __restrict__ x,
    const float* __restrict__ w1, const float* __restrict__ w3,
    const float* __restrict__ sw1, const float* __restrict__ sw3,
    const int* __restrict__ list, const int* __restrict__ cnt,
    float* __restrict__ hbuf) {
  int e = blockIdx.z;
  int n_e = cnt[e];
  int m0 = blockIdx.y * BM;
  if (m0 >= n_e) return;
  int h0 = blockIdx.x * BN;
  const float* W1 = (e < kE) ? (w1 + (size_t)e * kD * kH) : sw1;
  const float* W3 = (e < kE) ? (w3 + (size_t)e * kD * kH) : sw3;

  __shared__ __align__(16) float xs[2][BM][XPAD];
  __shared__ __align__(16) float w1s[2][BK][BN];
  __shared__ __align__(16) float w3s[2][BK][BN];
  __shared__ int   slts[BM];
  __shared__ int   toks[BM];

  int tid = threadIdx.x;
  if (tid < BM) {
    int p = m0 + tid;
    int s = (p < n_e) ? list[e * kN + p] : -1;
    slts[tid] = s;
    toks[tid] = (s >= 0) ? (s / kSLOTS) : -1;
  }
  __syncthreads();

  // per-thread async-copy slots (each thread moves 2 X rows-of-16B + 2+2 W)
  int rXA = tid >> 3,           cXA = (tid & 7) * 4;
  int rXB = (tid + 256) >> 3,   cXB = (tid & 7) * 4;
  int rWa = tid >> 4,           cWa = (tid & 15) * 4;
  int rWb = (tid + 256) >> 4,   cWb = (tid & 15) * 4;
  int tokA = toks[rXA]; if (tokA < 0) tokA = 0;   // clamp: pad rows load row 0,
  int tokB = toks[rXB]; if (tokB < 0) tokB = 0;   // results discarded at store
  const float* gXA  = x  + (size_t)tokA * kD + cXA;
  const float* gXB  = x  + (size_t)tokB * kD + cXB;
  const float* gW1a = W1 + (size_t)rWa * kH + h0 + cWa;
  const float* gW1b = W1 + (size_t)rWb * kH + h0 + cWb;
  const float* gW3a = W3 + (size_t)rWa * kH + h0 + cWa;
  const float* gW3b = W3 + (size_t)rWb * kH + h0 + cWb;
  unsigned int lXA[2]  = { lds_addr(&xs[0][rXA][cXA]),  lds_addr(&xs[1][rXA][cXA]) };
  unsigned int lXB[2]  = { lds_addr(&xs[0][rXB][cXB]),  lds_addr(&xs[1][rXB][cXB]) };
  unsigned int lW1a[2] = { lds_addr(&w1s[0][rWa][cWa]), lds_addr(&w1s[1][rWa][cWa]) };
  unsigned int lW1b[2] = { lds_addr(&w1s[0][rWb][cWb]), lds_addr(&w1s[1][rWb][cWb]) };
  unsigned int lW3a[2] = { lds_addr(&w3s[0][rWa][cWa]), lds_addr(&w3s[1][rWa][cWa]) };
  unsigned int lW3b[2] = { lds_addr(&w3s[0][rWb][cWb]), lds_addr(&w3s[1][rWb][cWb]) };

  auto issue = [&](int k0, int p) {
    async_copy_b128(lXA[p],  gXA + k0);
    async_copy_b128(lXB[p],  gXB + k0);
    async_copy_b128(lW1a[p], gW1a + (size_t)k0 * kH);
    async_copy_b128(lW1b[p], gW1b + (size_t)k0 * kH);
    async_copy_b128(lW3a[p], gW3a + (size_t)k0 * kH);
    async_copy_b128(lW3b[p], gW3b + (size_t)k0 * kH);
  };

  int wave = tid >> 5, lane = tid & 31;
  int r  = wave & 3;            // m-subtile
  int c0 = (wave >> 2) * 2;     // n-subtile pair
  int lh = lane >> 4;
  int l16 = lane & 15;

  v8f ag0 = {}, ag1 = {}, au0 = {}, au1 = {};

  issue(0, 0);
  for (int kt = 0; kt < NT; ++kt) {
    int p = kt & 1;
    issue(((kt + 1) & (NT - 1)) * BK, p ^ 1);   // prefetch next chunk (wraps)
    WAIT_ASYNC(0x6);                            // this wave's current chunk done
    __syncthreads();                            // everyone's chunk resident
#pragma unroll
    for (int kk = 0; kk < BK; kk += 4) {
      int ka = kk + lh * 2;
      v2f a = *(const v2f*)(&xs[p][r * 16 + l16][ka]);
      v2f b10, b11, b30, b31;
      b10.x = w1s[p][ka + 0][c0 * 16 + l16];        b10.y = w1s[p][ka + 1][c0 * 16 + l16];
      b11.x = w1s[p][ka + 0][(c0 + 1) * 16 + l16];  b11.y = w1s[p][ka + 1][(c0 + 1) * 16 + l16];
      b30.x = w3s[p][ka + 0][c0 * 16 + l16];        b30.y = w3s[p][ka + 1][c0 * 16 + l16];
      b31.x = w3s[p][ka + 0][(c0 + 1) * 16 + l16];  b31.y = w3s[p][ka + 1][(c0 + 1) * 16 + l16];
      ag0 = wmma4(a, b10, ag0);
      ag1 = wmma4(a, b11, ag1);
      au0 = wmma4(a, b30, au0);
      au1 = wmma4(a, b31, au1);
    }
    __syncthreads();   // done reading buf p before it is overwritten next iter
  }

#pragma unroll
  for (int v = 0; v < 8; ++v) {
    int mi = r * 16 + lh * 8 + v;      // C layout: lanes<16 -> M=v, lanes>=16 -> M=v+8
    int s = slts[mi];
    if (s >= 0) {
      float g0 = ag0[v], g1 = ag1[v];
      float hv0 = (g0 / (1.0f + __expf(-g0))) * au0[v];
      float hv1 = (g1 / (1.0f + __expf(-g1))) * au1[v];
      hbuf[(size_t)s * kH + h0 + c0 * 16 + l16] = hv0;
      hbuf[(size_t)s * kH + h0 + (c0 + 1) * 16 + l16] = hv1;
    }
  }
}

// ---------------- 3: gemm2  obuf[slot] = h[slot] @ W2 ----------------
#define NT2 (kH / BK)

__global__ __launch_bounds__(256) void moe_gemm2(
    const float* __restrict__ hbuf,
    const float* __restrict__ w2, const float* __restrict__ sw2,
    const int* __restrict__ list, const int* __restrict__ cnt,
    float* __restrict__ obuf) {
  int e = blockIdx.z;
  int n_e = cnt[e];
  int m0 = blockIdx.y * BM;
  if (m0 >= n_e) return;
  int d0 = blockIdx.x * BN;
  const float* W2 = (e < kE) ? (w2 + (size_t)e * kH * kD) : sw2;

  __shared__ __align__(16) float hs[2][BM][XPAD];
  __shared__ __align__(16) float w2s[2][BK][BN];
  __shared__ int   slts[BM];

  int tid = threadIdx.x;
  if (tid < BM) {
    int p = m0 + tid;
    slts[tid] = (p < n_e) ? list[e * kN + p] : -1;
  }
  __syncthreads();

  int rXA = tid >> 3,         cXA = (tid & 7) * 4;
  int rXB = (tid + 256) >> 3, cXB = (tid & 7) * 4;
  int rWa = tid >> 4,         cWa = (tid & 15) * 4;
  int rWb = (tid + 256) >> 4, cWb = (tid & 15) * 4;
  int sA = slts[rXA]; if (sA < 0) sA = 0;
  int sB = slts[rXB]; if (sB < 0) sB = 0;
  const float* gHA = hbuf + (size_t)sA * kH + cXA;
  const float* gHB = hbuf + (size_t)sB * kH + cXB;
  const float* gWa = W2 + (size_t)rWa * kD + d0 + cWa;
  const float* gWb = W2 + (size_t)rWb * kD + d0 + cWb;
  unsigned int lHA[2] = { lds_addr(&hs[0][rXA][cXA]),  lds_addr(&hs[1][rXA][cXA]) };
  unsigned int lHB[2] = { lds_addr(&hs[0][rXB][cXB]),  lds_addr(&hs[1][rXB][cXB]) };
  unsigned int lWa[2] = { lds_addr(&w2s[0][rWa][cWa]), lds_addr(&w2s[1][rWa][cWa]) };
  unsigned int lWb[2] = { lds_addr(&w2s[0][rWb][cWb]), lds_addr(&w2s[1][rWb][cWb]) };

  auto issue = [&](int k0, int p) {
    async_copy_b128(lHA[p], gHA + k0);
    async_copy_b128(lHB[p], gHB + k0);
    async_copy_b128(lWa[p], gWa + (size_t)k0 * kD);
    async_copy_b128(lWb[p], gWb + (size_t)k0 * kD);
  };

  int wave = tid >> 5, lane = tid & 31;
  int r  = wave & 3;
  int c0 = (wave >> 2) * 2;
  int lh = lane >> 4;
  int l16 = lane & 15;

  v8f a0 = {}, a1 = {};

  issue(0, 0);
  for (int kt = 0; kt < NT2; ++kt) {
    int p = kt & 1;
    issue(((kt + 1) & (NT2 - 1)) * BK, p ^ 1);
    WAIT_ASYNC(0x4);
    __syncthreads();
#pragma unroll
    for (int kk = 0; kk < BK; kk += 4) {
      int ka = kk + lh * 2;
      v2f a = *(const v2f*)(&hs[p][r * 16 + l16][ka]);
      v2f b0, b1;
      b0.x = w2s[p][ka + 0][c0 * 16 + l16];       b0.y = w2s[p][ka + 1][c0 * 16 + l16];
      b1.x = w2s[p][ka + 0][(c0 + 1) * 16 + l16]; b1.y = w2s[p][ka + 1][(c0 + 1) * 16 + l16];
      a0 = wmma4(a, b0, a0);
      a1 = wmma4(a, b1, a1);
    }
    __syncthreads();
  }

#pragma unroll
  for (int v = 0; v < 8; ++v) {
    int mi = r * 16 + lh * 8 + v;
    int s = slts[mi];
    if (s >= 0) {
      obuf[(size_t)s * kD + d0 + c0 * 16 + l16] = a0[v];
      obuf[(size_t)s * kD + d0 + (c0 + 1) * 16 + l16] = a1[v];
    }
  }
}

// ---------------- 4: combine slots -> output ----------------
__global__ __launch_bounds__(256) void moe_combine(
    const float* __restrict__ obuf, const float* __restrict__ coef,
    float* __restrict__ out) {
  int i = blockIdx.x * 256 + threadIdx.x;   // over kN*kD exactly
  int t = i >> 10;                          // / kD
  int d = i & (kD - 1);
  float acc = coef[t * kSLOTS + 0] * obuf[(size_t)(t * kSLOTS + 0) * kD + d]
            + coef[t * kSLOTS + 1] * obuf[(size_t)(t * kSLOTS + 1) * kD + d]
            + coef[t * kSLOTS + 2] * obuf[(size_t)(t * kSLOTS + 2) * kD + d];
  out[i] = acc;
}

// ---------------- 5: aux losses (deterministic tree reduce) ----------------
__global__ __launch_bounds__(256) void moe_finalize(
    const float* __restrict__ probsBuf, const float* __restrict__ lse2Buf,
    const int* __restrict__ cnt, float* __restrict__ losses) {
  __shared__ float red[256];
  int tid = threadIdx.x;
  float local[kE + 1];
#pragma unroll
  for (int q = 0; q <= kE; ++q) local[q] = 0.0f;
  for (int t = tid; t < kN; t += 256) {
#pragma unroll
    for (int e = 0; e < kE; ++e) local[e] += probsBuf[(size_t)t * kE + e];
    local[kE] += lse2Buf[t];
  }
  float results[kE + 1];
  for (int q = 0; q <= kE; ++q) {
    red[tid] = local[q];
    __syncthreads();
    for (int s2 = 128; s2 > 0; s2 >>= 1) {
      if (tid < s2) red[tid] += red[tid + s2];
      __syncthreads();
    }
    if (tid == 0) results[q] = red[0];
    __syncthreads();
  }
  if (tid == 0) {
    float lb = 0.0f;
#pragma unroll
    for (int e = 0; e < kE; ++e) lb += (float)cnt[e] * results[e];
    lb *= (8.0f * 0.01f) / (16.0f * (float)kN * 2.0f);  // E*LB_W/(LAYERS*N*K)
    float z = (results[kE] / (float)kN) * 0.001f;       // mean(lse^2)*Z_W
    losses[0] = lb;
    losses[1] = z;
  }
}

// ---------------- launcher ----------------
extern "C" void kernel_launch(void* const* d_in, const int* in_sizes, int n_in,
                              void* d_out, int out_size, void* d_ws, size_t ws_size,
                              hipStream_t stream) {
  (void)in_sizes; (void)n_in; (void)out_size; (void)ws_size;
  const float* x   = (const float*)d_in[0];
  const float* wr  = (const float*)d_in[1];
  const float* w1  = (const float*)d_in[2];
  const float* w3  = (const float*)d_in[3];
  const float* w2  = (const float*)d_in[4];
  const float* sw1 = (const float*)d_in[5];
  const float* sw3 = (const float*)d_in[6];
  const float* sw2 = (const float*)d_in[7];
  float* out = (float*)d_out;

  char* ws = (char*)d_ws;
  float* coef     = (float*)(ws + OFF_COEF);
  int*   list     = (int*)(ws + OFF_LIST);
  int*   cnt      = (int*)(ws + OFF_CNT);
  float* probsBuf = (float*)(ws + OFF_PROBS);
  float* lse2     = (float*)(ws + OFF_LSE2);
  float* hbuf     = (float*)(ws + OFF_HBUF);
  float* obuf     = (float*)(ws + OFF_OBUF);

  moe_init_counts<<<1, 32, 0, stream>>>(cnt);
  moe_router<<<kN / 8, 256, 0, stream>>>(x, wr, list, cnt, coef, probsBuf, lse2);
  moe_gemm1<<<dim3(kH / BN, kN / BM, kE + 1), 256, 0, stream>>>(
      x, w1, w3, sw1, sw3, list, cnt, hbuf);
  moe_gemm2<<<dim3(kD / BN, kN / BM, kE + 1), 256, 0, stream>>>(
      hbuf, w2, sw2, list, cnt, obuf);
  moe_combine<<<(kN * kD) / 256, 256, 0, stream>>>(obuf, coef, out);
  moe_finalize<<<1, 256, 0, stream>>>(probsBuf, lse2, cnt, out + (size_t)kN * kD);
}